// mLSTMCell_31044023616496
// MI455X (gfx1250) — compile-verified
//
#include <hip/hip_runtime.h>
#include <math.h>

typedef __attribute__((ext_vector_type(2))) float v2f;
typedef __attribute__((ext_vector_type(4))) float v4f;
typedef __attribute__((ext_vector_type(8))) float v8f;

#define B_  2
#define S_  2048
#define E_  512
#define NH_ 8
#define DH_ 64

#define KV_STRIDE 68   // 16x64 K/V tile rows padded: frag reads spread banks (4*ln)
#define SC_STRIDE 18   // score tile rows padded: 8B-aligned b64 reads, bank spread
#define USE_ASYNC_STAGE 1

__device__ __forceinline__ float logsig(float x) {
    // log sigmoid(x) = min(x,0) - log1p(exp(-|x|))
    return fminf(x, 0.0f) - log1pf(__expf(-fabsf(x)));
}

// ---------------------------------------------------------------------------
// Kernel 1: gate pre-activations ig/fg[b,nh,s] = dot(concat(q,k,v)[b,s,:], W[nh]) + b
// ---------------------------------------------------------------------------
__global__ void gates_kernel(const float* __restrict__ q,
                             const float* __restrict__ k,
                             const float* __restrict__ v,
                             const float* __restrict__ Wi,
                             const float* __restrict__ bi,
                             const float* __restrict__ Wf,
                             const float* __restrict__ bf,
                             float* __restrict__ ig,
                             float* __restrict__ fg) {
    __shared__ float gin[3 * E_];
    const int bs  = blockIdx.x;            // b*S + s
    const int b   = bs / S_;
    const int s   = bs % S_;
    const int tid = threadIdx.x;

    for (int e = tid; e < E_; e += 256) {
        gin[e]          = q[(size_t)bs * E_ + e];
        gin[E_ + e]     = k[(size_t)bs * E_ + e];
        gin[2 * E_ + e] = v[(size_t)bs * E_ + e];
    }
    __syncthreads();

    const int hg   = tid >> 4;   // 0..15 : (head, gate)
    const int p    = tid & 15;
    const int head = hg >> 1;
    const int gate = hg & 1;
    const float* W = (gate ? Wf : Wi) + head * 3 * E_;

    float acc = 0.0f;
    for (int e = p; e < 3 * E_; e += 16) acc += gin[e] * W[e];
    #pragma unroll
    for (int off = 1; off < 16; off <<= 1) acc += __shfl_xor(acc, off, 32);

    if (p == 0) {
        float bias = gate ? bf[head] : bi[head];
        float* dst = gate ? fg : ig;
        dst[((size_t)(b * NH_ + head)) * S_ + s] = acc + bias;
    }
}

// ---------------------------------------------------------------------------
// Kernel 2: per-(b,nh) scan: cs=cumsum(logsig(fg)), m=ig-cs, M=runmax(m),
// nexp=exp(-(cs+M)). One wave32 per (b,nh), chunked shfl scans with carry.
// ---------------------------------------------------------------------------
__global__ void scan_kernel(const float* __restrict__ ig,
                            const float* __restrict__ fg,
                            float* __restrict__ mS,
                            float* __restrict__ MS,
                            float* __restrict__ nexpS) {
    const int bh      = blockIdx.x;
    const int lane    = threadIdx.x;
    const size_t base = (size_t)bh * S_;

    float carry_cs = 0.0f;
    float carry_M  = -3.0e38f;
    for (int s0 = 0; s0 < S_; s0 += 32) {
        float f  = fg[base + s0 + lane];
        float iv = ig[base + s0 + lane];
        float lf = logsig(f);

        float cs = lf;                       // inclusive +scan
        #pragma unroll
        for (int off = 1; off < 32; off <<= 1) {
            float t = __shfl_up(cs, off, 32);
            if (lane >= off) cs += t;
        }
        cs += carry_cs;

        float m = iv - cs;

        float mm = m;                        // inclusive max-scan
        #pragma unroll
        for (int off = 1; off < 32; off <<= 1) {
            float t = __shfl_up(mm, off, 32);
            if (lane >= off) mm = fmaxf(mm, t);
        }
        float M = fmaxf(carry_M, mm);

        mS[base + s0 + lane]    = m;
        MS[base + s0 + lane]    = M;
        nexpS[base + s0 + lane] = __expf(-(cs + M));

        carry_cs = __shfl(cs, 31, 32);
        carry_M  = __shfl(M, 31, 32);
    }
}

// ---------------------------------------------------------------------------
// Kernel 3: causal flash mLSTM with fp32 WMMA + double-buffered async K/V
// staging into LDS. One block = 8 waves = 128 rows of one (b,nh); all waves
// share each staged 16-key K/V tile. Per tile per wave: 16 WMMA (QK^T, K=64),
// decay/mask/rowsum, LDS transpose, 16 WMMA (S@V into 4 d-chunks).
// ---------------------------------------------------------------------------
__global__ void __launch_bounds__(256) flash_kernel(
        const float* __restrict__ q,
        const float* __restrict__ k,
        const float* __restrict__ v,
        const float* __restrict__ mS,
        const float* __restrict__ MS,
        const float* __restrict__ nexpS,
        float* __restrict__ out) {
    __shared__ float kbuf[2][16 * KV_STRIDE];
    __shared__ float vbuf[2][16 * KV_STRIDE];
    __shared__ float sbuf[8][16 * SC_STRIDE];

    const int tid  = threadIdx.x;
    const int wv   = tid >> 5;
    const int lane = tid & 31;
    const int half = lane >> 4;
    const int ln   = lane & 15;

    const int blk     = blockIdx.x;
    const int hb      = blk >> 4;            // (b,nh) flat  (S/128 == 16 blocks each)
    const int blkrt   = blk & 15;            // which 128-row slab
    const int b       = hb >> 3;
    const int nh      = hb & 7;
    const int rowbase = blkrt * 128 + wv * 16;   // this wave's 16-row tile
    const int T       = blkrt * 8 + 8;           // column tiles for this block

    // staging role: thread -> (row 0..15, 4-col group) of the 16x64 tile
    const int srow = tid >> 4;
    const int sc4  = (tid & 15) * 4;

    const float* qh  = q + (size_t)b * S_ * E_ + nh * DH_;
    const float* kh  = k + (size_t)b * S_ * E_ + nh * DH_;
    const float* vh  = v + (size_t)b * S_ * E_ + nh * DH_;
    const float* mh  = mS    + (size_t)(b * NH_ + nh) * S_;
    const float* Mh  = MS    + (size_t)(b * NH_ + nh) * S_;
    const float* nxh = nexpS + (size_t)(b * NH_ + nh) * S_;

    // Q A-fragments (16x4 f32: lane=row, VGPR pair=K chunk by lane half), *DH^-0.5
    v2f aq[16];
    {
        const float* qr = qh + (size_t)(rowbase + ln) * E_;
        #pragma unroll
        for (int kk = 0; kk < 16; ++kk) {
            v2f t = *(const v2f*)(qr + kk * 4 + 2 * half);
            aq[kk] = t * 0.125f;
        }
    }
    float Mr[8];
    #pragma unroll
    for (int r = 0; r < 8; ++r) Mr[r] = Mh[rowbase + r + 8 * half];

    v8f vzero = {};
    v8f hacc[4];
    #pragma unroll
    for (int c = 0; c < 4; ++c) hacc[c] = vzero;
    float rs[8];
    #pragma unroll
    for (int r = 0; r < 8; ++r) rs[r] = 0.0f;

#if USE_ASYNC_STAGE
    // async gather of one K + one V 16B chunk per thread into LDS (ASYNCcnt +2/wave)
    auto stage = [&](int t, int bi) {
        const int cb = t * 16;
        unsigned long long gk =
            (unsigned long long)(uintptr_t)(kh + (size_t)(cb + srow) * E_ + sc4);
        unsigned long long gv =
            (unsigned long long)(uintptr_t)(vh + (size_t)(cb + srow) * E_ + sc4);
        unsigned int lk = (unsigned int)(uintptr_t)&kbuf[bi][srow * KV_STRIDE + sc4];
        unsigned int lv = (unsigned int)(uintptr_t)&vbuf[bi][srow * KV_STRIDE + sc4];
        asm volatile("global_load_async_to_lds_b128 %0, %1, off"
                     :: "v"(lk), "v"(gk) : "memory");
        asm volatile("global_load_async_to_lds_b128 %0, %1, off"
                     :: "v"(lv), "v"(gv) : "memory");
    };
    stage(0, 0);
#endif

    for (int t = 0; t < T; ++t) {
        const int bi      = t & 1;
        const int colbase = t * 16;

        __syncthreads();   // buffer bi^1 consumers (iter t-1) are done
#if USE_ASYNC_STAGE
        if (t + 1 < T) {
            stage(t + 1, bi ^ 1);
            asm volatile("s_wait_asynccnt 0x2" ::: "memory");  // tile t retired (in order)
        } else {
            asm volatile("s_wait_asynccnt 0x0" ::: "memory");
        }
#else
        {
            v4f kk4 = *(const v4f*)(kh + (size_t)(colbase + srow) * E_ + sc4);
            v4f vv4 = *(const v4f*)(vh + (size_t)(colbase + srow) * E_ + sc4);
            *(v4f*)&kbuf[bi][srow * KV_STRIDE + sc4] = kk4;
            *(v4f*)&vbuf[bi][srow * KV_STRIDE + sc4] = vv4;
        }
#endif
        __syncthreads();   // staged tile t visible to all waves

        if (colbase <= rowbase) {   // wave-uniform causal guard (EXEC stays all-1)
            // QK^T: 16 chained fp32 WMMAs, B-fragments from LDS
            v8f acc = vzero;
            #pragma unroll
            for (int kk = 0; kk < 16; ++kk) {
                v2f bk = *(const v2f*)&kbuf[bi][ln * KV_STRIDE + kk * 4 + 2 * half];
                acc = __builtin_amdgcn_wmma_f32_16x16x4_f32(
                    false, aq[kk], false, bk, (short)0, acc, false, false);
            }

            // decay + causal mask + rowsum; stash scores (acc layout) to LDS
            const float mj = mh[colbase + ln];
            #pragma unroll
            for (int r = 0; r < 8; ++r) {
                const int ir = rowbase + r + 8 * half;
                const int j  = colbase + ln;
                float pfac = (j <= ir) ? __expf(mj - Mr[r]) : 0.0f;
                float sc = acc[r] * pfac;
                rs[r] += sc;
                sbuf[wv][(r + 8 * half) * SC_STRIDE + ln] = sc;
            }
            asm volatile("s_wait_dscnt 0" ::: "memory");

            // h += scores @ V : A = transposed scores from LDS, B = V tile from LDS
            #pragma unroll
            for (int jj = 0; jj < 4; ++jj) {
                v2f as = *(const v2f*)&sbuf[wv][ln * SC_STRIDE + jj * 4 + 2 * half];
                const int j0 = jj * 4 + 2 * half;   // tile-local key row
                #pragma unroll
                for (int c = 0; c < 4; ++c) {
                    v2f bv;
                    bv.x = vbuf[bi][j0 * KV_STRIDE + c * 16 + ln];
                    bv.y = vbuf[bi][(j0 + 1) * KV_STRIDE + c * 16 + ln];
                    hacc[c] = __builtin_amdgcn_wmma_f32_16x16x4_f32(
                        false, as, false, bv, (short)0, hacc[c], false, false);
                }
            }
        }
    }

    // rowsum reduce across the 16 column-lanes (halves are independent rows)
    #pragma unroll
    for (int r = 0; r < 8; ++r) {
        float t = rs[r];
        #pragma unroll
        for (int off = 1; off < 16; off <<= 1) t += __shfl_xor(t, off, 32);
        rs[r] = t;
    }

    // normalizer + per-row layernorm over DH=64, store
    #pragma unroll
    for (int r = 0; r < 8; ++r) {
        const int ir = rowbase + r + 8 * half;
        float nx  = nxh[ir];
        float inv = 1.0f / (fmaxf(fabsf(rs[r]), nx) + 1e-6f);
        float h0 = hacc[0][r] * inv;
        float h1 = hacc[1][r] * inv;
        float h2 = hacc[2][r] * inv;
        float h3 = hacc[3][r] * inv;
        float s1 = h0 + h1 + h2 + h3;
        float s2 = h0 * h0 + h1 * h1 + h2 * h2 + h3 * h3;
        #pragma unroll
        for (int off = 1; off < 16; off <<= 1) {
            s1 += __shfl_xor(s1, off, 32);
            s2 += __shfl_xor(s2, off, 32);
        }
        float mean = s1 * (1.0f / 64.0f);
        float var  = s2 * (1.0f / 64.0f) - mean * mean;
        float rstd = rsqrtf(var + 1e-5f);
        float* orow = out + (size_t)(b * S_ + ir) * E_ + nh * DH_;
        orow[0 * 16 + ln] = (h0 - mean) * rstd;
        orow[1 * 16 + ln] = (h1 - mean) * rstd;
        orow[2 * 16 + ln] = (h2 - mean) * rstd;
        orow[3 * 16 + ln] = (h3 - mean) * rstd;
    }
}

extern "C" void kernel_launch(void* const* d_in, const int* in_sizes, int n_in,
                              void* d_out, int out_size, void* d_ws, size_t ws_size,
                              hipStream_t stream) {
    const float* q  = (const float*)d_in[0];
    const float* k  = (const float*)d_in[1];
    const float* v  = (const float*)d_in[2];
    const float* Wi = (const float*)d_in[3];
    const float* bi = (const float*)d_in[4];
    const float* Wf = (const float*)d_in[5];
    const float* bf = (const float*)d_in[6];

    // workspace: 5 float arrays of (B,NH,S) = 32768 elements each (640 KB)
    float* ws    = (float*)d_ws;
    float* ig    = ws;
    float* fg    = ws + 32768;
    float* mS    = ws + 65536;
    float* MS    = ws + 98304;
    float* nexpS = ws + 131072;

    gates_kernel<<<B_ * S_, 256, 0, stream>>>(q, k, v, Wi, bi, Wf, bf, ig, fg);
    scan_kernel<<<B_ * NH_, 32, 0, stream>>>(ig, fg, mS, MS, nexpS);
    flash_kernel<<<B_ * NH_ * (S_ / 128), 256, 0, stream>>>(
        q, k, v, mS, MS, nexpS, (float*)d_out);
}